// FusedTensorProductOp3_69295002354278
// MI455X (gfx1250) — compile-verified
//
#include <hip/hip_runtime.h>
#include <math.h>

typedef float v2f __attribute__((ext_vector_type(2)));
typedef float v8f __attribute__((ext_vector_type(8)));
typedef int   v4i __attribute__((ext_vector_type(4)));

#define NFEAT 1152   // 128*1 + 128*3 + 128*5
#define WAVES 8
#define BUFSZ 1168   // 1152 data + 8 zero-pad (reads c>=9) + 8 dump (writes l15>=9)
#define ZPAD  1152   // zero region base
#define DPAD  1160   // dump region base
#define STSZ  1040   // padded T: max idx (11*9+15)*9+8 = 1034

typedef __attribute__((address_space(1))) v4i v4i_g;   // global int4
typedef __attribute__((address_space(3))) v4i v4i_l;   // LDS int4

// ---------------------------------------------------------------------------
// Init kernel: replicate np.random.RandomState(42).randn(...) path coefficients
// and scatter them into T[i][n][j] (i=input idx 0..8, n=output idx, j=x1 idx).
// ---------------------------------------------------------------------------
struct MTState {
  unsigned* mt;
  int mti;
  __device__ unsigned next() {
    if (mti >= 624) {
      for (int k = 0; k < 624; ++k) {
        unsigned y = (mt[k] & 0x80000000u) | (mt[(k + 1) % 624] & 0x7fffffffu);
        mt[k] = mt[(k + 397) % 624] ^ (y >> 1) ^ ((y & 1u) ? 2567483615u : 0u);
      }
      mti = 0;
    }
    unsigned y = mt[mti++];
    y ^= y >> 11;
    y ^= (y << 7) & 2636928640u;
    y ^= (y << 15) & 4022730752u;
    y ^= y >> 18;
    return y;
  }
  __device__ double rk_double() {
    unsigned a = next() >> 5, b = next() >> 6;
    return (a * 67108864.0 + b) / 9007199254740992.0;
  }
};

struct GaussState {
  MTState* m;
  double cache;
  int has;
  __device__ double next() {
    if (has) { has = 0; return cache; }
    double u, v, r2;
    do {
      u = 2.0 * m->rk_double() - 1.0;
      v = 2.0 * m->rk_double() - 1.0;
      r2 = u * u + v * v;
    } while (r2 >= 1.0 || r2 == 0.0);
    double f = sqrt(-2.0 * log(r2) / r2);
    cache = f * u;
    has = 1;
    return f * v;
  }
};

__global__ void ftp_init_T(float* __restrict__ T) {
  __shared__ unsigned mt[624];
  if (threadIdx.x != 0) return;
  for (int i = 0; i < 729; ++i) T[i] = 0.0f;
  unsigned sd = 42u;
  for (int i = 0; i < 624; ++i) {
    mt[i] = sd;
    sd = 1812433253u * (sd ^ (sd >> 30)) + (unsigned)(i + 1);
  }
  MTState m{mt, 624};
  GaussState g{&m, 0.0, 0};
  const int D_[3] = {1, 3, 5};
  const int O_[3] = {0, 1, 4};
  for (int l1 = 0; l1 < 3; ++l1)
    for (int l2 = 0; l2 < 3; ++l2) {
      int lo = l1 - l2; if (lo < 0) lo = -lo;
      int hi = l1 + l2; if (hi > 2) hi = 2;
      for (int l3 = lo; l3 <= hi; ++l3) {
        double sc = sqrt((double)(D_[l1] * D_[l2]));
        for (int i = 0; i < D_[l1]; ++i)
          for (int j = 0; j < D_[l2]; ++j)
            for (int k = 0; k < D_[l3]; ++k) {
              float c32 = (float)g.next();           // .astype(float32)
              float val = (float)((double)c32 / sc); // C /= sqrt(d1*d2)
              T[(O_[l1] + i) * 81 + (O_[l3] + k) * 9 + (O_[l2] + j)] = val;
            }
      }
    }
}

// ---------------------------------------------------------------------------
// Helpers
// ---------------------------------------------------------------------------
template <int N>
__device__ __forceinline__ void wait_asynccnt() {
#if __has_builtin(__builtin_amdgcn_s_wait_asynccnt)
  asm volatile("" ::: "memory");
  __builtin_amdgcn_s_wait_asynccnt(N);
  asm volatile("" ::: "memory");
#else
  asm volatile("s_wait_asynccnt %0" ::"n"(N) : "memory");
#endif
}

// Async (ASYNCcnt) copy of one 4608B sample row into this wave's LDS buffer.
__device__ __forceinline__ void prefetch_row(const float* __restrict__ g,
                                             float* lbuf, int lane) {
#if __has_builtin(__builtin_amdgcn_global_load_async_to_lds_b128)
#pragma unroll
  for (int q = 0; q < 9; ++q) {
    const float* gp = g + (lane + 32 * q) * 4;
    float* lp = lbuf + (lane + 32 * q) * 4;
    __builtin_amdgcn_global_load_async_to_lds_b128((v4i_g*)gp, (v4i_l*)lp, 0, 0);
  }
#else
  const float4* src = (const float4*)g;
  float4* dst = (float4*)lbuf;
#pragma unroll
  for (int q = 0; q < 9; ++q) dst[lane + 32 * q] = src[lane + 32 * q];
#endif
}

// Branch-free index into the e3nn channel layout; c>=9 -> zero pad.
__device__ __forceinline__ int a_idx(int row, int c) {
  return (c == 0) ? row
       : (c < 4)  ? (128 + row * 3 + (c - 1))
       : (c < 9)  ? (512 + row * 5 + (c - 4))
                  : (ZPAD + (c - 9));
}

// Branch-free output index; c>=9 -> dump pad (write-only scratch).
__device__ __forceinline__ int o_idx(int row, int c) {
  return (c == 0) ? row
       : (c < 4)  ? (128 + row * 3 + (c - 1))
       : (c < 9)  ? (512 + row * 5 + (c - 4))
                  : (DPAD + (c - 9));
}

// ---------------------------------------------------------------------------
// Main kernel: per sample, Out(128x9) = A(128x9) @ W(9x9), W = T . b
// ---------------------------------------------------------------------------
__global__ __launch_bounds__(256) void ftp_main(const float* __restrict__ x0,
                                                const float* __restrict__ x1,
                                                float* __restrict__ out,
                                                int Z,
                                                const float* __restrict__ T) {
  __shared__ __align__(16) float sT[STSZ];
  __shared__ __align__(16) float sBuf[WAVES][2][BUFSZ];

  const int tid = threadIdx.x;
  for (int i = tid; i < STSZ; i += 256) sT[i] = (i < 729) ? T[i] : 0.0f;
  __syncthreads();

  const int lane = tid & 31;
  const int wv = tid >> 5;
  const int half = lane >> 4;   // 0: lanes 0-15, 1: lanes 16-31
  const int l15 = lane & 15;

  // zero the read-pad of both ping-pong buffers (dump pad can hold garbage)
  if (lane < 8) {
    sBuf[wv][0][ZPAD + lane] = 0.0f;
    sBuf[wv][1][ZPAD + lane] = 0.0f;
  }

  const int gw = blockIdx.x * WAVES + wv;
  const int nw = gridDim.x * WAVES;

  int z = gw;
  int pb = 0;
  if (z < Z) prefetch_row(x0 + (size_t)z * NFEAT, sBuf[wv][0], lane);

  for (; z < Z; z += nw) {
    float* s = sBuf[wv][pb];
    const int zn = z + nw;
    if (zn < Z) {
      prefetch_row(x0 + (size_t)zn * NFEAT, sBuf[wv][pb ^ 1], lane);
      wait_asynccnt<9>();   // in-order: current buffer's 9 ops have landed
    } else {
      wait_asynccnt<0>();
    }

    // ---- load b (9 floats) ----
    const float* bp = x1 + (size_t)z * 9;
    float bv[9];
#pragma unroll
    for (int j = 0; j < 9; ++j) bv[j] = bp[j];

    // ---- build W in B-operand layout: wreg[2*st+v] = W[4*st+2*half+v][l15] ----
    // (sT is zero-padded, so i>=9 gives 0; l15>=9 gives junk columns that go to the dump pad — no guards)
    float wreg[6];
#pragma unroll
    for (int sv = 0; sv < 6; ++sv) {
      const int st = sv >> 1, v = sv & 1;
      const int i = 4 * st + 2 * half + v;
      const float* tr = &sT[(i * 9 + l15) * 9];
      float acc = 0.0f;
#pragma unroll
      for (int j = 0; j < 9; ++j) acc = fmaf(tr[j], bv[j], acc);
      wreg[sv] = acc;
    }

    // ---- 8 tiles of 16 channel rows: D(16x16) += A(16x4) x B(4x16), 3 K-steps ----
#pragma unroll
    for (int t = 0; t < 8; ++t) {
      const int row = t * 16 + l15;
      float areg[6];
#pragma unroll
      for (int sv = 0; sv < 6; ++sv) {
        const int st = sv >> 1, v = sv & 1;
        areg[sv] = s[a_idx(row, 4 * st + 2 * half + v)];
      }
      v8f acc = {};
#pragma unroll
      for (int st = 0; st < 3; ++st) {
        v2f av = {areg[2 * st], areg[2 * st + 1]};
        v2f bw = {wreg[2 * st], wreg[2 * st + 1]};
        acc = __builtin_amdgcn_wmma_f32_16x16x4_f32(false, av, false, bw,
                                                    (short)0, acc, false, false);
      }
      // D layout: VGPR r -> (M = r + 8*half, N = l15); l15>=9 goes to dump pad
#pragma unroll
      for (int r = 0; r < 8; ++r) {
        const int orow = t * 16 + half * 8 + r;
        s[o_idx(orow, l15)] = acc[r];
      }
    }

    // ---- coalesced b128 store of the finished row (fire-and-forget) ----
    float4* dsto = (float4*)(out + (size_t)z * NFEAT);
    const float4* s4 = (const float4*)s;
#pragma unroll
    for (int q = 0; q < 9; ++q) dsto[lane + 32 * q] = s4[lane + 32 * q];

    pb ^= 1;
  }
}

extern "C" void kernel_launch(void* const* d_in, const int* in_sizes, int n_in,
                              void* d_out, int out_size, void* d_ws, size_t ws_size,
                              hipStream_t stream) {
  (void)n_in; (void)out_size; (void)ws_size;
  const float* x0 = (const float*)d_in[0];
  const float* x1 = (const float*)d_in[1];
  float* out = (float*)d_out;
  float* T = (float*)d_ws;

  const int Z = in_sizes[0] / NFEAT;

  ftp_init_T<<<1, 32, 0, stream>>>(T);

  int blocks = (Z + WAVES - 1) / WAVES;
  if (blocks > 2048) blocks = 2048;
  if (blocks < 1) blocks = 1;
  ftp_main<<<blocks, 256, 0, stream>>>(x0, x1, out, Z, T);
}